// GNN_41042707480955
// MI455X (gfx1250) — compile-verified
//
#include <hip/hip_runtime.h>

#define DIM     128
#define NNODES  50000
#define NEDGES  800000
#define LSTR    132   // padded LDS row stride (floats): 132 % 64 = 4 -> conflict-free column reads

typedef __attribute__((ext_vector_type(2))) float v2f;
typedef __attribute__((ext_vector_type(8))) float v8f;

// Guaranteed hardware f32 atomic add (GLOBAL_ATOMIC_ADD_F32, no return -> STOREcnt).
// Inline asm so it can never silently become a CAS loop.
__device__ __forceinline__ void atomic_add_gl(float* p, float v) {
#if defined(__gfx1250__)
    asm volatile("global_atomic_add_f32 %0, %1, off"
                 :
                 : "v"(p), "v"(v)
                 : "memory");
#else
    unsafeAtomicAdd(p, v);
#endif
}

// ---------------- utility: zero a float buffer (n multiple of 4) ----------------
__global__ void __launch_bounds__(256) k_zero(float* __restrict__ p, int n) {
    int i4 = (blockIdx.x * 256 + threadIdx.x) * 4;
    if (i4 < n) *(float4*)(p + i4) = make_float4(0.f, 0.f, 0.f, 0.f);
}

// ---------------- degree count: deg[dst[e]] += 1.0 ----------------
__global__ void __launch_bounds__(256) k_deg(const int* __restrict__ dst,
                                             float* __restrict__ deg, int ne) {
    int i = blockIdx.x * 256 + threadIdx.x;
    if (i < ne) atomic_add_gl(&deg[dst[i]], 1.0f);
}

__global__ void __launch_bounds__(256) k_invdeg(float* __restrict__ deg, int n) {
    int i = blockIdx.x * 256 + threadIdx.x;
    if (i < n) deg[i] = 1.0f / fmaxf(deg[i], 1.0f);
}

// ---------------- scatter-add: agg[dst[e]] += h[src[e]], one edge per wave32 ----------------
__global__ void __launch_bounds__(256) k_scatter(const float* __restrict__ h,
                                                 const int* __restrict__ src,
                                                 const int* __restrict__ dst,
                                                 float* __restrict__ agg, int ne) {
    const int gtid = blockIdx.x * 256 + threadIdx.x;
    const int e    = gtid >> 5;          // one edge per wave
    const int lane = threadIdx.x & 31;
    if (e >= ne) return;
    const int s = __builtin_amdgcn_readfirstlane(src[e]);   // lane-uniform -> SGPR addressing
    const int d = __builtin_amdgcn_readfirstlane(dst[e]);
    // 32 lanes x float4 = full 512B row, coalesced gather
    const float4 v = *(const float4*)(h + (size_t)s * DIM + lane * 4);
    float* p = agg + (size_t)d * DIM + lane * 4;
    atomic_add_gl(p + 0, v.x);
    atomic_add_gl(p + 1, v.y);
    atomic_add_gl(p + 2, v.z);
    atomic_add_gl(p + 3, v.w);
}

// ---------------- fused SAGE layer GEMM:
// out = relu?( h @ Ws + (agg*inv_deg) @ Wn + bias )
// block = 16 rows x 128 cols; 8 waves, each wave owns a 16x16 tile via V_WMMA_F32_16X16X4_F32
__global__ void __launch_bounds__(256) k_sage_gemm(const float* __restrict__ h,
                                                   const float* __restrict__ agg,
                                                   const float* __restrict__ invdeg,
                                                   const float* __restrict__ Ws,
                                                   const float* __restrict__ Wn,
                                                   const float* __restrict__ bias,
                                                   float* __restrict__ out,
                                                   int relu) {
    __shared__ float sh[16 * LSTR];   // h tile          (16 x 128, stride 132)
    __shared__ float sa[16 * LSTR];   // agg*invdeg tile

    const int m0  = blockIdx.x * 16;  // 50000 = 3125 * 16, no tail
    const int tid = threadIdx.x;

    // ---- cooperative staging: thread t -> row t/16, 8 floats at col (t%16)*8 ----
    {
        const int    r    = tid >> 4;
        const int    c    = (tid & 15) * 8;
        const size_t base = (size_t)(m0 + r) * DIM + c;
        float4 h0 = *(const float4*)(h + base);
        float4 h1 = *(const float4*)(h + base + 4);
        float4 a0 = *(const float4*)(agg + base);
        float4 a1 = *(const float4*)(agg + base + 4);
        const float idg = invdeg[m0 + r];
        a0.x *= idg; a0.y *= idg; a0.z *= idg; a0.w *= idg;
        a1.x *= idg; a1.y *= idg; a1.z *= idg; a1.w *= idg;
        float* dh = &sh[r * LSTR + c];
        float* da = &sa[r * LSTR + c];
        *(float4*)(dh)     = h0;
        *(float4*)(dh + 4) = h1;
        *(float4*)(da)     = a0;
        *(float4*)(da + 4) = a1;
    }
    __syncthreads();

    const int wave = tid >> 5;
    const int lane = tid & 31;
    const int hi   = lane >> 4;      // half-wave select (K split / M split)
    const int lr   = lane & 15;
    const int n0   = wave * 16;
    const int col  = n0 + lr;

    v8f acc = {};

    // ISA 16x16x4 f32 fragment layout:
    //  A: lanes 0-15 hold (M=lr, K=k0..k0+1), lanes 16-31 hold (M=lr, K=k0+2..k0+3)
    //  B: lanes 0-15 hold (K=k0..k0+1, N=lr), lanes 16-31 hold (K=k0+2..k0+3, N=lr)
#pragma unroll 8
    for (int k0 = 0; k0 < DIM; k0 += 4) {
        const int ka = k0 + hi * 2;
        v2f aS = *(const v2f*)&sh[lr * LSTR + ka];
        v2f aN = *(const v2f*)&sa[lr * LSTR + ka];
        v2f bS, bN;
        bS.x = Ws[(size_t)ka * DIM + col];
        bS.y = Ws[(size_t)(ka + 1) * DIM + col];
        bN.x = Wn[(size_t)ka * DIM + col];
        bN.y = Wn[(size_t)(ka + 1) * DIM + col];
        acc = __builtin_amdgcn_wmma_f32_16x16x4_f32(false, aS, false, bS,
                                                    (short)0, acc, false, false);
        acc = __builtin_amdgcn_wmma_f32_16x16x4_f32(false, aN, false, bN,
                                                    (short)0, acc, false, false);
    }

    // ---- epilogue: D VGPR r -> (M = hi*8 + r, N = col) ----
    const float bv    = bias[col];
    const int   rbase = m0 + hi * 8;
#pragma unroll
    for (int r = 0; r < 8; ++r) {
        float v = acc[r] + bv;
        if (relu) v = fmaxf(v, 0.f);
        out[(size_t)(rbase + r) * DIM + col] = v;
    }
}

// ---------------- host-side orchestration ----------------
extern "C" void kernel_launch(void* const* d_in, const int* in_sizes, int n_in,
                              void* d_out, int out_size, void* d_ws, size_t ws_size,
                              hipStream_t stream) {
    const float* x   = (const float*)d_in[0];   // [50000,128]
    const int*   src = (const int*)  d_in[1];   // [800000]
    const int*   dst = (const int*)  d_in[2];   // [800000]
    const float* Ws  = (const float*)d_in[3];   // [3,128,128]
    const float* Wn  = (const float*)d_in[4];   // [3,128,128]
    const float* b   = (const float*)d_in[5];   // [3,128]
    float* out = (float*)d_out;                 // [50000,128]

    const size_t feat = (size_t)NNODES * DIM;
    float* agg = (float*)d_ws;                  // 6.4M floats
    float* hA  = agg + feat;                    // 6.4M floats (layer-0 output)
    float* idg = hA + feat;                     // 50k floats  (deg -> inv_deg in place)

    // --- degrees: deg = segsum(1, dst); inv_deg = 1/max(deg,1) ---
    k_zero  <<<(NNODES / 4 + 255) / 256, 256, 0, stream>>>(idg, NNODES);
    k_deg   <<<(NEDGES + 255) / 256, 256, 0, stream>>>(dst, idg, NEDGES);
    k_invdeg<<<(NNODES + 255) / 256, 256, 0, stream>>>(idg, NNODES);

    const int scatter_blocks = (NEDGES + 7) / 8;   // 8 waves (edges) per block
    const int gemm_blocks    = NNODES / 16;        // 3125
    const int zero_blocks    = (int)(feat / 4 + 255) / 256;

    for (int l = 0; l < 3; ++l) {
        const float* hin  = (l == 0) ? x : ((l == 1) ? hA : out);
        float*       hout = (l == 0) ? hA : out;   // layer 2 is in-place per 16-row block
        k_zero   <<<zero_blocks, 256, 0, stream>>>(agg, (int)feat);
        k_scatter<<<scatter_blocks, 256, 0, stream>>>(hin, src, dst, agg, NEDGES);
        k_sage_gemm<<<gemm_blocks, 256, 0, stream>>>(
            hin, agg, idg,
            Ws + (size_t)l * DIM * DIM,
            Wn + (size_t)l * DIM * DIM,
            b + (size_t)l * DIM,
            hout, (l < 2) ? 1 : 0);
    }
}